// TorchsparseUpsample_63496796504724
// MI455X (gfx1250) — compile-verified
//
#include <hip/hip_runtime.h>
#include <stdint.h>

typedef unsigned long long u64;
typedef unsigned int u32;
typedef int   v4i __attribute__((ext_vector_type(4)));
typedef float v4f __attribute__((ext_vector_type(4)));

#define EMPTY_KEY 0xFFFFFFFFFFFFFFFFULL

#define AS1 __attribute__((address_space(1)))
#define AS3 __attribute__((address_space(3)))

// Global-side pointer: produce an AS1 v4i* via integer round-trip (flat
// address == global address on AMDGPU). LDS-side: direct addrspace cast.
#define GPTR(p) ((AS1 v4i*)(uintptr_t)(const void*)(p))
#define LPTR(p) ((AS3 v4i*)(p))

#if defined(__has_builtin)
#if __has_builtin(__builtin_amdgcn_global_load_async_to_lds_b128) && \
    __has_builtin(__builtin_amdgcn_global_store_async_from_lds_b128) && \
    __has_builtin(__builtin_amdgcn_s_wait_asynccnt)
#define USE_ASYNC_LDS 1
#else
#define USE_ASYNC_LDS 0
#endif
#else
#define USE_ASYNC_LDS 0
#endif

__device__ __forceinline__ u64 make_key(int b, int x, int y, int z) {
    return ((u64)(u32)b << 48) | ((u64)(u32)x << 32) |
           ((u64)(u32)y << 16) |  (u64)(u32)z;
}

__device__ __forceinline__ u32 hash_key(u64 k, u32 mask) {
    k *= 0x9E3779B97F4A7C15ULL;           // Fibonacci mix
    return (u32)(k >> 32) & mask;         // take well-mixed high bits
}

// ---------------------------------------------------------------- init table
__global__ void init_table_kernel(u64* __restrict__ keys, int T) {
    int i = blockIdx.x * blockDim.x + threadIdx.x;
    if (i < T) keys[i] = EMPTY_KEY;
}

// ------------------------------------------------------------------- insert
__global__ void insert_kernel(const int4* __restrict__ in_coords,
                              u64* __restrict__ keys, int* __restrict__ vals,
                              int N, u32 mask) {
    int i = blockIdx.x * blockDim.x + threadIdx.x;
    if (i >= N) return;
    int4 c = in_coords[i];                               // [b, x, y, z]
    u64 key = make_key(c.x, c.y, c.z, c.w);
    u32 slot = hash_key(key, mask);
    for (;;) {
        u64 prev = atomicCAS(&keys[slot], EMPTY_KEY, key);
        if (prev == EMPTY_KEY) { vals[slot] = i; return; }
        slot = (slot + 1) & mask;                        // keys are unique
    }
}

// ------------------------------------------------------------------- gather
// Wave handles 4 output rows per iteration. Each row = 32 floats = 8 float4
// chunks; 8 lanes per row, 1 chunk per lane -> one 512B coalesced b128
// load + store per wave-iteration.
__global__ __launch_bounds__(256) void gather_kernel(
        const int4*   __restrict__ out_coords,
        const float4* __restrict__ feats,
        const u64*    __restrict__ keys,
        const int*    __restrict__ vals,
        float4*       __restrict__ out,
        int M, u32 mask) {
#if USE_ASYNC_LDS
    __shared__ float4 stage[8][32];       // 8 waves/block * 32 lanes * 16B
    const int w = threadIdx.x >> 5;
#endif
    const int lane  = threadIdx.x & 31;
    const int rown  = lane >> 3;          // 0..3  : row within wave
    const int chunk = lane & 7;           // 0..7  : float4 chunk within row
    const int wave   = (int)((blockIdx.x * blockDim.x + threadIdx.x) >> 5);
    const int nwaves = (int)((gridDim.x * blockDim.x) >> 5);

    for (int base = wave * 4; base < M; base += nwaves * 4) {
        int m = base + rown;
        // Prefetch next iteration's coord stream (global_prefetch_b8).
        int nm = m + nwaves * 4;
        if (nm < M) __builtin_prefetch(&out_coords[nm], 0, 0);

        if (m < M) {
            int4 oc = out_coords[m];                     // [b, x, y, z] fine voxel
            u64 key = make_key(oc.x, oc.y >> 1, oc.z >> 1, oc.w >> 1);
            u32 slot = hash_key(key, mask);
            while (keys[slot] != key) slot = (slot + 1) & mask;
            int p = vals[slot];                          // parent row index

#if USE_ASYNC_LDS
            // Payload never touches VGPRs: global -> LDS -> global via the
            // gfx1250 async path, tracked with ASYNCcnt.
            __builtin_amdgcn_s_wait_asynccnt(0);   // prior store drained; LDS reusable
            __builtin_amdgcn_global_load_async_to_lds_b128(
                GPTR(feats + (p * 8 + chunk)), LPTR(&stage[w][lane]), 0, 0);
            __builtin_amdgcn_s_wait_asynccnt(0);   // load landed in LDS
            __builtin_amdgcn_global_store_async_from_lds_b128(
                GPTR(out + (m * 8 + chunk)), LPTR(&stage[w][lane]), 0, 0);
#else
            // Fallback: direct VGPR copy, non-temporal store for write-once
            // output. Native clang vector type keeps the builtin happy on
            // both host and device passes.
            v4f v = *(const v4f*)(feats + (p * 8 + chunk));   // L2-resident reuse
            __builtin_nontemporal_store(v, (v4f*)(out + (m * 8 + chunk)));
#endif
        }
    }
    // s_endpgm implicitly waits all counters (incl. ASYNCcnt).
}

// -------------------------------------------------------------------- launch
extern "C" void kernel_launch(void* const* d_in, const int* in_sizes, int n_in,
                              void* d_out, int out_size, void* d_ws, size_t ws_size,
                              hipStream_t stream) {
    const int N = in_sizes[0] / 4;   // coarse voxels
    const int M = in_sizes[2] / 4;   // fine voxels (8N)

    const int4*   in_coords  = (const int4*)d_in[0];
    const float4* feats      = (const float4*)d_in[1];
    const int4*   out_coords = (const int4*)d_in[2];
    float4*       out        = (float4*)d_out;

    // Hash table size: power of two >= 2N (load factor <= 0.5), shrink if ws
    // is too small. 12 bytes/slot: 8B key + 4B value.
    u32 T = 1;
    while ((long long)T < 2LL * N) T <<= 1;
    while ((size_t)T * 12 > ws_size && T > 2) T >>= 1;
    u64* keys = (u64*)d_ws;
    int* vals = (int*)((char*)d_ws + (size_t)T * 8);
    u32 mask = T - 1;

    init_table_kernel<<<(int)((T + 255) / 256), 256, 0, stream>>>(keys, (int)T);
    insert_kernel<<<(N + 255) / 256, 256, 0, stream>>>(in_coords, keys, vals, N, mask);

    // 256 threads = 8 waves; 4 rows/wave-iter -> 32 rows per block sweep.
    int blocks = (M + 31) / 32;
    gather_kernel<<<blocks, 256, 0, stream>>>(out_coords, feats, keys, vals, out,
                                              M, mask);
}